// BiLSTM_7842610282762
// MI455X (gfx1250) — compile-verified
//
#include <hip/hip_runtime.h>
#include <hip/hip_bf16.h>

typedef float v2f __attribute__((ext_vector_type(2)));
typedef float v8f __attribute__((ext_vector_type(8)));

#define TT 128
#define BB 64
#define EE 32
#define HH 8
#define DD 40
#define VV 32000
#define NROW (TT * BB)          // 8192 rows of total_h
#define NTILE (VV / 16)         // 2000 vocab tiles

// ---------------------------------------------------------------------------
// Kernel 1: one wave per (batch element); run twice (dir=0 fwd, dir=1 bwd).
// Lanes 0..2 compute f/i/o gates, lanes 3..10 compute the 8 WC columns.
// Writes the PRE-step h (matching jax.lax.scan collecting old h) into
// total_h[(t*B+b)*16 + dir*8 + j].
// ---------------------------------------------------------------------------
__global__ __launch_bounds__(32) void BiLSTM_lstm_kernel(
    const int* __restrict__ x, const float* __restrict__ emb,
    const float* __restrict__ Wf, const float* __restrict__ bf,
    const float* __restrict__ Wi, const float* __restrict__ bi,
    const float* __restrict__ WC, const float* __restrict__ bC,
    const float* __restrict__ Wo, const float* __restrict__ bo,
    int dir, float* __restrict__ hout)
{
    const int b    = blockIdx.x;
    const int lane = threadIdx.x;

    __shared__ float hs[DD];      // [h(8) | e(32)]
    __shared__ float gates[3];    // f, i, o

    // Preload this lane's weight column into registers.
    float w[DD];
    float bias = 0.f;
    {
        const float* base;
        int str;
        int j = (lane >= 3) ? ((lane - 3) & 7) : 0;
        if      (lane == 0) { base = Wf;     str = 1;  bias = bf[0]; }
        else if (lane == 1) { base = Wi;     str = 1;  bias = bi[0]; }
        else if (lane == 2) { base = Wo;     str = 1;  bias = bo[0]; }
        else                { base = WC + j; str = HH; bias = bC[j]; }
        #pragma unroll
        for (int d = 0; d < DD; ++d) w[d] = base[d * str];
    }

    if (lane < HH) hs[lane] = 0.f;
    float Creg = 0.f;
    float ct   = 0.f;
    __syncthreads();

    for (int s = 0; s < TT; ++s) {
        const int t = dir ? (TT - 1 - s) : s;

        // Emit pre-step hidden state.
        if (lane < HH)
            hout[(t * BB + b) * (2 * HH) + dir * HH + lane] = hs[lane];

        // Gather embedding row for this timestep.
        const int idx = x[t * BB + b];
        hs[HH + lane] = emb[idx * EE + lane];
        __syncthreads();

        if (lane < 11) {
            float acc = bias;
            #pragma unroll
            for (int d = 0; d < DD; ++d) acc += hs[d] * w[d];
            if (lane < 3) gates[lane] = 1.f / (1.f + __expf(-acc));
            else          ct = tanhf(acc);
        }
        __syncthreads();

        if (lane >= 3 && lane < 11) {
            const float f  = gates[0];
            const float ig = gates[1];
            const float o  = gates[2];
            const float Cn = f * Creg + ig + ct;   // NOTE: ref is f*C + i + Ct
            Creg = Cn;
            hs[lane - 3] = o * tanhf(Cn);
        }
        __syncthreads();
    }
}

// ---------------------------------------------------------------------------
// WMMA helper: one 16x16 logits tile via 4 chained V_WMMA_F32_16X16X4_F32.
// A layout (16x4 f32): lane m=lane&15 -> M row; hi=lane>>4; vgpr j -> K = 4*kk + 2*hi + j
// B layout (4x16 f32): mirrored; C/D: vgpr r -> rows r (lanes 0-15) / r+8 (lanes 16-31)
// ---------------------------------------------------------------------------
__device__ __forceinline__ v8f logits_tile(const v2f a[4],
                                           const float* __restrict__ Wout,
                                           int n0, int m, int hi)
{
    v8f c = {};
    #pragma unroll
    for (int kk = 0; kk < 4; ++kk) {
        const int k0 = kk * 4 + hi * 2;
        v2f bmat;
        bmat.x = Wout[(k0 + 0) * VV + n0 + m];
        bmat.y = Wout[(k0 + 1) * VV + n0 + m];
        c = __builtin_amdgcn_wmma_f32_16x16x4_f32(
                /*neg_a=*/false, a[kk], /*neg_b=*/false, bmat,
                /*c_mod=*/(short)0, c, /*reuse_a=*/false, /*reuse_b=*/false);
    }
    return c;
}

__device__ __forceinline__ void load_a_tile(v2f a[4], const float* __restrict__ th,
                                            int rt, int m, int hi)
{
    const float* arow = th + (rt * 16 + m) * 16;
    #pragma unroll
    for (int kk = 0; kk < 4; ++kk) {
        a[kk].x = arow[kk * 4 + hi * 2 + 0];
        a[kk].y = arow[kk * 4 + hi * 2 + 1];
    }
}

// ---------------------------------------------------------------------------
// Kernel 2: per-row max + logsumexp over all 32000 logits (flash-style online
// accumulation; logits never hit HBM). 1 block (8 waves) per 16-row tile;
// waves stride across the 2000 vocab tiles; LDS combine at the end.
// ---------------------------------------------------------------------------
__global__ __launch_bounds__(256) void BiLSTM_lse_kernel(
    const float* __restrict__ th, const float* __restrict__ Wout,
    const float* __restrict__ bout, float* __restrict__ lse)
{
    const int rt   = blockIdx.x;
    const int tid  = threadIdx.x;
    const int wave = tid >> 5;
    const int lane = tid & 31;
    const int m    = lane & 15;
    const int hi   = lane >> 4;

    v2f a[4];
    load_a_tile(a, th, rt, m, hi);

    float mr[8], sr[8];
    #pragma unroll
    for (int r = 0; r < 8; ++r) { mr[r] = -1e30f; sr[r] = 0.f; }

    for (int nt = wave; nt < NTILE; nt += 8) {
        const int n0 = nt * 16;
        const v8f c  = logits_tile(a, Wout, n0, m, hi);
        const float bv = bout[n0 + m];
        #pragma unroll
        for (int r = 0; r < 8; ++r) {
            const float v  = c[r] + bv;
            const float mo = mr[r];
            const float mn = fmaxf(mo, v);
            sr[r] = sr[r] * __expf(mo - mn) + __expf(v - mn);
            mr[r] = mn;
        }
    }

    __shared__ float Lm[256][8];
    __shared__ float Ls[256][8];
    #pragma unroll
    for (int r = 0; r < 8; ++r) { Lm[tid][r] = mr[r]; Ls[tid][r] = sr[r]; }
    __syncthreads();

    if (tid < 16) {                 // tid == row within tile (r + 8*hh)
        const int r  = tid & 7;
        const int hh = tid >> 3;
        float M = -1e30f, S = 0.f;
        for (int w = 0; w < 8; ++w) {
            for (int l = 0; l < 16; ++l) {
                const int src = w * 32 + hh * 16 + l;
                const float pm = Lm[src][r];
                const float ps = Ls[src][r];
                const float mn = fmaxf(M, pm);
                S = S * __expf(M - mn) + ps * __expf(pm - mn);
                M = mn;
            }
        }
        lse[rt * 16 + tid] = M + __logf(S);
    }
}

// ---------------------------------------------------------------------------
// Kernel 3: recompute logits (Wout is L2-resident, recompute beats spilling
// 1 GB) and write log_softmax with non-temporal stores. One wave per
// (row-tile, 16 vocab-tile chunk).
// ---------------------------------------------------------------------------
__global__ __launch_bounds__(32) void BiLSTM_out_kernel(
    const float* __restrict__ th, const float* __restrict__ Wout,
    const float* __restrict__ bout, const float* __restrict__ lse,
    float* __restrict__ out)
{
    const int rt    = blockIdx.x;
    const int chunk = blockIdx.y;      // 125 chunks * 16 tiles = 2000
    const int lane  = threadIdx.x;
    const int m     = lane & 15;
    const int hi    = lane >> 4;

    v2f a[4];
    load_a_tile(a, th, rt, m, hi);

    float cr[8];
    #pragma unroll
    for (int r = 0; r < 8; ++r) cr[r] = lse[rt * 16 + hi * 8 + r];

    for (int i = 0; i < 16; ++i) {
        const int n0 = (chunk * 16 + i) * 16;
        const v8f c  = logits_tile(a, Wout, n0, m, hi);
        const float bv = bout[n0 + m];
        #pragma unroll
        for (int r = 0; r < 8; ++r) {
            const int row = rt * 16 + hi * 8 + r;
            __builtin_nontemporal_store(c[r] + bv - cr[r], &out[row * VV + n0 + m]);
        }
    }
}

// ---------------------------------------------------------------------------
extern "C" void kernel_launch(void* const* d_in, const int* in_sizes, int n_in,
                              void* d_out, int out_size, void* d_ws, size_t ws_size,
                              hipStream_t stream)
{
    const int*   x    = (const int*)  d_in[0];
    const float* emb  = (const float*)d_in[1];
    const float* Wf1  = (const float*)d_in[2];
    const float* bf1  = (const float*)d_in[3];
    const float* Wi1  = (const float*)d_in[4];
    const float* bi1  = (const float*)d_in[5];
    const float* WC1  = (const float*)d_in[6];
    const float* bC1  = (const float*)d_in[7];
    const float* Wo1  = (const float*)d_in[8];
    const float* bo1  = (const float*)d_in[9];
    const float* Wf2  = (const float*)d_in[10];
    const float* bf2  = (const float*)d_in[11];
    const float* Wi2  = (const float*)d_in[12];
    const float* bi2  = (const float*)d_in[13];
    const float* WC2  = (const float*)d_in[14];
    const float* bC2  = (const float*)d_in[15];
    const float* Wo2  = (const float*)d_in[16];
    const float* bo2  = (const float*)d_in[17];
    const float* Wout = (const float*)d_in[18];
    const float* bout = (const float*)d_in[19];

    float* ws_h   = (float*)d_ws;            // NROW * 16 floats = 512 KB
    float* ws_lse = ws_h + NROW * 16;        // NROW floats       = 32 KB

    BiLSTM_lstm_kernel<<<BB, 32, 0, stream>>>(x, emb, Wf1, bf1, Wi1, bi1,
                                              WC1, bC1, Wo1, bo1, 0, ws_h);
    BiLSTM_lstm_kernel<<<BB, 32, 0, stream>>>(x, emb, Wf2, bf2, Wi2, bi2,
                                              WC2, bC2, Wo2, bo2, 1, ws_h);

    BiLSTM_lse_kernel<<<NROW / 16, 256, 0, stream>>>(ws_h, Wout, bout, ws_lse);

    dim3 grid(NROW / 16, 125);
    BiLSTM_out_kernel<<<grid, 32, 0, stream>>>(ws_h, Wout, bout, ws_lse,
                                               (float*)d_out);
}